// GLIF_81329500717443
// MI455X (gfx1250) — compile-verified
//
#include <hip/hip_runtime.h>
#include <hip/hip_bf16.h>

// GLIF forward for N=8192.
// Dominant cost: w @ I_additive (fp32 GEMV, 256 MB of w traffic -> HBM bound,
// ~11.5us floor at 23.3 TB/s). Strategy:
//   * 512 workgroups, one per 16-row output block; 1024 threads = 32 waves.
//   * Each wave owns a 256-column K segment and runs 64 x V_WMMA_F32_16X16X4_F32.
//   * A-fragment per ISA layout = contiguous float2 per lane (full cacheline
//     consumption across consecutive K-steps via L0).
//   * B-fragment = I_additive broadcast across N columns (L2-resident 32KB).
//   * Deterministic cross-wave reduction in LDS, fused GLIF elementwise epilogue.

typedef __attribute__((ext_vector_type(2))) float v2f;
typedef __attribute__((ext_vector_type(8))) float v8f;

#define NN     8192
#define WAVES  32            // waves per workgroup
#define KSEG   (NN / WAVES)  // 256 columns per wave
#define KSTEP  4             // K per WMMA

// GLIF constants
#define F_V          0.15f
#define DELTA_V      12.0f

__launch_bounds__(1024)
__global__ void glif_fused_wmma(const float* __restrict__ x_in,
                                const float* __restrict__ w,
                                const float* __restrict__ i_add,
                                const float* __restrict__ v_in,
                                const float* __restrict__ theta_s,
                                const float* __restrict__ theta_v,
                                const float* __restrict__ tau_m,
                                const float* __restrict__ R_I,
                                const float* __restrict__ v_rest,
                                float* __restrict__ out_v,
                                float* __restrict__ out_sig)
{
    __shared__ float red[WAVES][17];   // [wave][row], padded to dodge bank conflicts

    const int tid  = threadIdx.x;
    const int wave = tid >> 5;
    const int lane = tid & 31;
    const int half = lane >> 4;        // 0: lanes 0-15, 1: lanes 16-31
    const int lr   = lane & 15;        // row index within the 16-row tile
    const int m0   = blockIdx.x << 4;  // first output row of this block

    const int kbeg = wave * KSEG;

    // A fragment (ISA 16x4 fp32 layout): lane<16 -> w[row][k..k+1],
    // lane>=16 -> w[row][k+2..k+3]  => one contiguous float2 per lane.
    const float* arow = w + (size_t)(m0 + lr) * NN + (kbeg + 2 * half);
    // B fragment: broadcast of I_additive along N; K split mirrors A halves.
    const float* bvec = i_add + (kbeg + 2 * half);

    v8f acc = {};
#pragma unroll 4
    for (int s = 0; s < KSEG / KSTEP; ++s) {
        v2f a = *(const v2f*)(arow + s * KSTEP);
        v2f b = *(const v2f*)(bvec + s * KSTEP);
        // D = A(16x4) * B(4x16) + C ; every D column is identical (B columns equal)
        acc = __builtin_amdgcn_wmma_f32_16x16x4_f32(
            /*neg_a=*/false, a, /*neg_b=*/false, b,
            /*c_mod=*/(short)0, acc, /*reuse_a=*/false, /*reuse_b=*/false);
    }

    // C/D layout: VGPR r -> (M=r, N=lane) for lanes 0-15, (M=8+r, N=lane-16)
    // for lanes 16-31. Column N=0 lives in lanes 0 (rows 0-7) and 16 (rows 8-15).
    if (lane == 0) {
#pragma unroll
        for (int r = 0; r < 8; ++r) red[wave][r] = acc[r];
    } else if (lane == 16) {
#pragma unroll
        for (int r = 0; r < 8; ++r) red[wave][8 + r] = acc[r];
    }
    __syncthreads();

    if (tid < 16) {
        const int m = m0 + tid;
        // Deterministic fixed-order reduction over the 32 K-segments.
        float dot = 0.0f;
#pragma unroll
        for (int wv = 0; wv < WAVES; ++wv) dot += red[wv][tid];

        const float I       = x_in[m] + dot;
        const float vm      = v_in[m];
        const float vr      = v_rest[m];
        const float v_next  = vm + (vr - vm + I * R_I[m]) / tau_m[m];
        const float thresh  = theta_s[m] + theta_v[m];
        const float sig     = 1.0f / (1.0f + __expf(-(v_next - thresh)));
        const float v_reset = vr + F_V * (vm - vr) - DELTA_V;
        const float v_new   = (v_next >= thresh) ? v_reset : v_next;

        out_v[m]   = v_new;
        out_sig[m] = sig;
    }
}

extern "C" void kernel_launch(void* const* d_in, const int* in_sizes, int n_in,
                              void* d_out, int out_size, void* d_ws, size_t ws_size,
                              hipStream_t stream) {
    (void)in_sizes; (void)n_in; (void)out_size; (void)d_ws; (void)ws_size;
    const float* x_in    = (const float*)d_in[0];
    const float* w       = (const float*)d_in[1];
    const float* i_add   = (const float*)d_in[2];
    const float* v       = (const float*)d_in[3];
    const float* theta_s = (const float*)d_in[4];
    const float* theta_v = (const float*)d_in[5];
    const float* tau_m   = (const float*)d_in[6];
    const float* R_I     = (const float*)d_in[7];
    const float* v_rest  = (const float*)d_in[8];

    float* out_v   = (float*)d_out;
    float* out_sig = (float*)d_out + NN;

    dim3 grid(NN / 16);   // 512 row-blocks
    dim3 block(1024);     // 32 waves
    glif_fused_wmma<<<grid, block, 0, stream>>>(x_in, w, i_add, v, theta_s,
                                                theta_v, tau_m, R_I, v_rest,
                                                out_v, out_sig);
}